// QModel_43885975830736
// MI455X (gfx1250) — compile-verified
//
#include <hip/hip_runtime.h>

// ---------------------------------------------------------------------------
// Quantum circuit -> one fixed 256x256 complex unitary M = Pg*K1*Pg*K0.
// probs = [ ||M x||^2 / ||x||^2 , 0 ].  Batch GEMM runs on f32 WMMA; the X
// tile is staged into LDS (in WMMA A-fragment order) with
// GLOBAL_LOAD_ASYNC_TO_LDS_B64 tracked on ASYNCcnt.
// ---------------------------------------------------------------------------

typedef float v2f __attribute__((ext_vector_type(2)));
typedef float v8f __attribute__((ext_vector_type(8)));

#define NB   16384
#define IMG  784
#define DIM  256

// float offsets inside workspace (total 1.5 MB)
#define W_OFF(l, p) (((l) * 2 + (p)) * 65536)   // W0re,W0im,W1re,W1im
#define MF_RE_OFF   (4 * 65536)                 // M packed in B-fragment order
#define MF_IM_OFF   (5 * 65536)

__device__ __forceinline__ v8f wmma_f32(v2f a, v2f b, v8f c) {
  // args: (neg_a, A, neg_b, B, c_mod, C, reuse_a, reuse_b)
  return __builtin_amdgcn_wmma_f32_16x16x4_f32(false, a, false, b, (short)0, c,
                                               false, false);
}

// ---------------------------------------------------------------------------
// Kernel 1: build W0, W1.  W_l[j,i] = K_l[g(i), j] = prod_w U_lw[bit(g(i)),bit(j)]
// g = composition of the 28 CNOTs (reversed application order on the index).
// ---------------------------------------------------------------------------
__global__ void build_w_kernel(const float* __restrict__ wts,
                               float* __restrict__ ws) {
  int gid = blockIdx.x * blockDim.x + threadIdx.x;  // 0 .. 131071
  int l   = gid >> 16;
  int idx = gid & 0xFFFF;
  int j   = idx >> 8;    // row of W (k index of the big GEMM)
  int i   = idx & 0xFF;  // column of W

  // permutation g(i): CNOT(c,t) on flat bits: bit pos of wire w is (7-w)
  int gi = i;
  for (int c = 6; c >= 0; --c)
    for (int t = 7; t > c; --t) {
      int cb = (gi >> (7 - c)) & 1;
      gi ^= cb << (7 - t);
    }

  // per-wire Rot unitaries U = RZ(c) RX(b) RZ(a)
  float ur[8][2][2], ui[8][2][2];
  for (int w = 0; w < 8; ++w) {
    float a = wts[(l * 9 + w) * 3 + 0];
    float b = wts[(l * 9 + w) * 3 + 1];
    float c = wts[(l * 9 + w) * 3 + 2];
    float sb, cb, sr, cr, sq, cq;
    __sincosf(0.5f * b, &sb, &cb);
    __sincosf(0.5f * (a + c), &sr, &cr);  // r = (a+c)/2
    __sincosf(0.5f * (a - c), &sq, &cq);  // q = (a-c)/2
    ur[w][0][0] =  cb * cr;  ui[w][0][0] = -cb * sr;   // cos(b/2) e^{-ir}
    ur[w][1][1] =  cb * cr;  ui[w][1][1] =  cb * sr;   // cos(b/2) e^{+ir}
    ur[w][0][1] =  sb * sq;  ui[w][0][1] = -sb * cq;   // -i sin(b/2) e^{+iq}
    ur[w][1][0] = -sb * sq;  ui[w][1][0] = -sb * cq;   // -i sin(b/2) e^{-iq}
  }

  float pr = 1.f, pi = 0.f;
  for (int w = 0; w < 8; ++w) {
    int rb = (gi >> (7 - w)) & 1;
    int cb = (j  >> (7 - w)) & 1;
    float re = ur[w][rb][cb], im = ui[w][rb][cb];
    float nr = pr * re - pi * im;
    float ni = pr * im + pi * re;
    pr = nr; pi = ni;
  }
  ws[W_OFF(l, 0) + j * DIM + i] = pr;
  ws[W_OFF(l, 1) + j * DIM + i] = pi;
}

// ---------------------------------------------------------------------------
// Kernel 2: M = W0 * W1 (complex 256^3 GEMM), one wave per 16x16 tile.
// Output written pre-swizzled into B-fragment order:
//   Mf[plane][ct][ks][lane][2] with lane = n + 16*((k>>1)&1), elem = k&1
// ---------------------------------------------------------------------------
__global__ void build_m_kernel(float* __restrict__ ws) {
  int tile = blockIdx.x;          // 0..255
  int jt   = (tile >> 4) * 16;    // row tile base
  int it   = tile & 15;           // col tile index
  int lane = threadIdx.x;         // single wave32
  int m    = lane & 15;
  int half = lane >> 4;
  int kk   = half * 2;

  const float* A_re = ws + W_OFF(0, 0);
  const float* A_im = ws + W_OFF(0, 1);
  const float* B_re = ws + W_OFF(1, 0);
  const float* B_im = ws + W_OFF(1, 1);

  v8f rr = {}, ii = {}, ri = {}, ir = {};
  for (int ks = 0; ks < 64; ++ks) {
    int k0 = ks * 4 + kk;
    v2f ar = *(const v2f*)(A_re + (jt + m) * DIM + k0);
    v2f ai = *(const v2f*)(A_im + (jt + m) * DIM + k0);
    v2f br, bi;
    br.x = B_re[(k0    ) * DIM + it * 16 + m];
    br.y = B_re[(k0 + 1) * DIM + it * 16 + m];
    bi.x = B_im[(k0    ) * DIM + it * 16 + m];
    bi.y = B_im[(k0 + 1) * DIM + it * 16 + m];
    rr = wmma_f32(ar, br, rr);
    ii = wmma_f32(ai, bi, ii);
    ri = wmma_f32(ar, bi, ri);
    ir = wmma_f32(ai, br, ir);
  }

  float* Mf_re = ws + MF_RE_OFF;
  float* Mf_im = ws + MF_IM_OFF;
  for (int r = 0; r < 8; ++r) {
    int k  = jt + r + 8 * half;                   // M row == batch-GEMM k
    int lp = m + 16 * ((k >> 1) & 1);
    int ei = ((it * 64 + (k >> 2)) * 32 + lp) * 2 + (k & 1);
    Mf_re[ei] = rr[r] - ii[r];                    // C_re = ArBr - AiBi
    Mf_im[ei] = ri[r] + ir[r];                    // C_im = ArBi + AiBr
  }
}

// ---------------------------------------------------------------------------
// Kernel 3: 32 batch rows per WG.  Stage X tile into LDS in A-fragment order
// via GLOBAL_LOAD_ASYNC_TO_LDS_B64 (each 8B chunk == one fragment slot),
// y = x * M via f32 WMMA (2 row-tiles x 2 col-tiles x re/im per wave),
// probs0 = sum|y|^2 / ||x||^2, probs1 = 0.
// ---------------------------------------------------------------------------
__global__ void __launch_bounds__(256, 2)
batch_probs_kernel(const float* __restrict__ x, const float* __restrict__ ws,
                   float* __restrict__ out) {
  __shared__ float Afrag[2][64][32][2];  // 32 KB, [rowtile][ks][lane][elem]
  __shared__ float norm2[32];
  __shared__ float rowsum[32];

  const int t    = threadIdx.x;
  const int row0 = blockIdx.x * 32;
  if (t < 32) { norm2[t] = 0.f; rowsum[t] = 0.f; }

  // ---- async staging: thread t covers row r = t>>3, 16 eight-byte chunks ----
  const int r     = t >> 3;              // local row 0..31
  const int rt    = r >> 4;              // row tile
  const int mm    = r & 15;              // row within tile
  const int qbase = (t & 7) * 16;        // first chunk (of 128 per row)
  const float* grow = x + (size_t)(row0 + r) * IMG;
  const unsigned lds_base = (unsigned)(uintptr_t)&Afrag[0][0][0][0];
#pragma unroll
  for (int q = 0; q < 16; ++q) {
    int c    = (qbase + q) * 2;          // even column -> 8B aligned
    int ks   = c >> 2;
    int lane = mm + (((c >> 1) & 1) << 4);
    unsigned lds = lds_base + (unsigned)(((rt * 64 + ks) * 32 + lane) * 8);
    unsigned long long ga = (unsigned long long)(uintptr_t)(grow + c);
    asm volatile("global_load_async_to_lds_b64 %0, %1, off"
                 :: "v"(lds), "v"(ga) : "memory");
  }
  asm volatile("s_wait_asynccnt 0x0" ::: "memory");
  __syncthreads();

  // ---- row norms (read staged data back from LDS) ----
  {
    float ss = 0.f;
#pragma unroll
    for (int q = 0; q < 16; ++q) {
      int c    = (qbase + q) * 2;
      int ks   = c >> 2;
      int lane = mm + (((c >> 1) & 1) << 4);
      v2f v = *(const v2f*)&Afrag[rt][ks][lane][0];
      ss += v.x * v.x + v.y * v.y;
    }
    atomicAdd(&norm2[r], ss);  // ds_add_f32
  }
  __syncthreads();

  // ---- GEMM: wave handles col-tiles 2w,2w+1 for both row tiles ----
  const int wave = t >> 5;
  const int lane = t & 31;
  const int half = lane >> 4;
  const v2f* Br0 = (const v2f*)(ws + MF_RE_OFF) + (wave * 2    ) * 64 * 32;
  const v2f* Bi0 = (const v2f*)(ws + MF_IM_OFF) + (wave * 2    ) * 64 * 32;
  const v2f* Br1 = (const v2f*)(ws + MF_RE_OFF) + (wave * 2 + 1) * 64 * 32;
  const v2f* Bi1 = (const v2f*)(ws + MF_IM_OFF) + (wave * 2 + 1) * 64 * 32;

  v8f cre00 = {}, cim00 = {}, cre01 = {}, cim01 = {};  // row tile 0
  v8f cre10 = {}, cim10 = {}, cre11 = {}, cim11 = {};  // row tile 1
#pragma unroll 4
  for (int ks = 0; ks < 64; ++ks) {
    v2f a0  = *(const v2f*)&Afrag[0][ks][lane][0];
    v2f a1  = *(const v2f*)&Afrag[1][ks][lane][0];
    v2f br0 = Br0[ks * 32 + lane];
    v2f bi0 = Bi0[ks * 32 + lane];
    v2f br1 = Br1[ks * 32 + lane];
    v2f bi1 = Bi1[ks * 32 + lane];
    cre00 = wmma_f32(a0, br0, cre00);
    cim00 = wmma_f32(a0, bi0, cim00);
    cre01 = wmma_f32(a0, br1, cre01);
    cim01 = wmma_f32(a0, bi1, cim01);
    cre10 = wmma_f32(a1, br0, cre10);
    cim10 = wmma_f32(a1, bi0, cim10);
    cre11 = wmma_f32(a1, br1, cre11);
    cim11 = wmma_f32(a1, bi1, cim11);
  }

  // ---- |y|^2 row reduction (VGPR rr holds row rr (+8 upper half-wave)) ----
#pragma unroll
  for (int rr = 0; rr < 8; ++rr) {
    float v0 = cre00[rr] * cre00[rr] + cim00[rr] * cim00[rr] +
               cre01[rr] * cre01[rr] + cim01[rr] * cim01[rr];
    float v1 = cre10[rr] * cre10[rr] + cim10[rr] * cim10[rr] +
               cre11[rr] * cre11[rr] + cim11[rr] * cim11[rr];
    atomicAdd(&rowsum[rr + 8 * half], v0);       // ds_add_f32
    atomicAdd(&rowsum[16 + rr + 8 * half], v1);
  }
  __syncthreads();

  if (t < 32) {
    float p = rowsum[t] / norm2[t];
    out[(size_t)(row0 + t) * 2 + 0] = p;
    out[(size_t)(row0 + t) * 2 + 1] = 0.f;  // wire 8 never acted on
  }
}

// ---------------------------------------------------------------------------
extern "C" void kernel_launch(void* const* d_in, const int* in_sizes, int n_in,
                              void* d_out, int out_size, void* d_ws,
                              size_t ws_size, hipStream_t stream) {
  const float* x   = (const float*)d_in[0];  // (16384,1,28,28) f32
  const float* wts = (const float*)d_in[1];  // (2,9,3) f32
  float* out = (float*)d_out;                // (16384,2) f32
  float* ws  = (float*)d_ws;                 // needs 1.5 MB

  build_w_kernel<<<512, 256, 0, stream>>>(wts, ws);
  build_m_kernel<<<256, 32, 0, stream>>>(ws);
  batch_probs_kernel<<<NB / 32, 256, 0, stream>>>(x, ws, out);
}